// UniPhyBlock_22093311770921
// MI455X (gfx1250) — compile-verified
//
#include <hip/hip_runtime.h>
#include <math.h>

// ---------------------------------------------------------------------------
// UniPhyBlock for MI455X (gfx1250, wave32).  All GEMM-shaped stages use
// V_WMMA_F32_16X16X4_F32 (fp32 matrix pipe -> bit-compatible precision with
// the fp32 reference).  Shapes: B=2, T=16, D=64, H=W=64.
// ---------------------------------------------------------------------------

#define BB   2
#define TT   16
#define DD   64
#define HH   64
#define WW   64
#define HWN  (HH * WW)          // 4096
#define NPOS (BB * TT * HWN)    // 131072 total (b,t,h,w) positions
#define C2   (2 * DD)           // 128

typedef float v2f __attribute__((ext_vector_type(2)));
typedef float v8f __attribute__((ext_vector_type(8)));

// D = A(16x4) * B(4x16) + C(16x16), fp32.
// 8-arg form: (neg_a, A, neg_b, B, c_mod, C, reuse_a, reuse_b)
#define WMMA_F32(a, b, c) \
  __builtin_amdgcn_wmma_f32_16x16x4_f32(false, (a), false, (b), (short)0, (c), false, false)

// ---------------------------------------------------------------------------
// Kernel 1: LayerNorm over 128 channels (concat(re,im)), planar output.
// x_re/x_im: (B,T,D,H,W).  xn: (B,T,128,H,W)
// ---------------------------------------------------------------------------
__global__ void k_ln(const float* __restrict__ x_re, const float* __restrict__ x_im,
                     const float* __restrict__ g_s,  const float* __restrict__ b_s,
                     float* __restrict__ xn) {
  int p  = blockIdx.x * 256 + threadIdx.x;     // 0..NPOS-1
  int bt = p >> 12;
  int sp = p & 4095;
  const float* xr = x_re + (size_t)bt * DD * HWN + sp;
  const float* xi = x_im + (size_t)bt * DD * HWN + sp;
  float s = 0.f, ss = 0.f;
  for (int d = 0; d < DD; ++d) {
    float a = xr[(size_t)d * HWN];
    float b = xi[(size_t)d * HWN];
    s += a + b; ss += a * a + b * b;
  }
  float m   = s * (1.f / 128.f);
  float var = ss * (1.f / 128.f) - m * m;
  float rs  = rsqrtf(var + 1e-5f);
  float* o = xn + (size_t)bt * C2 * HWN + sp;
  for (int d = 0; d < DD; ++d) {
    float a = xr[(size_t)d * HWN];
    float b = xi[(size_t)d * HWN];
    o[(size_t)d * HWN]        = (a - m) * rs * g_s[d]      + b_s[d];
    o[(size_t)(DD + d) * HWN] = (b - m) * rs * g_s[DD + d] + b_s[DD + d];
  }
}

// ---------------------------------------------------------------------------
// Kernel 2: 3x3 SAME conv 128->128 as implicit GEMM (K = 9*128 = 1152),
// fused bias + residual-with-original-x.  Output x_new channel-last complex.
//   xn:  (B,T,128,H,W) planar   Wc: (3,3,128,128)
//   x_new_re/im: (B,T,H*W,64)
// Block: 256 thr (8 waves).  Tile: M = 16 consecutive w in one row h,
// N-tile per wave (8*16 = 128 out channels).
// ---------------------------------------------------------------------------
__global__ void k_conv(const float* __restrict__ xn,  const float* __restrict__ Wc,
                       const float* __restrict__ bc,
                       const float* __restrict__ x_re, const float* __restrict__ x_im,
                       float* __restrict__ xnew_re,   float* __restrict__ xnew_im) {
  __shared__ float tile[C2][3][18];            // 27.6 KB halo tile
  int tidx = blockIdx.x;                       // 0..8191
  int bt   = tidx >> 8;                        // 256 tiles per image
  int tl   = tidx & 255;
  int h    = tl >> 2;                          // 64 rows
  int w0   = (tl & 3) * 16;                    // 4 col-tiles

  const float* img = xn + (size_t)bt * C2 * HWN;
  for (int idx = threadIdx.x; idx < C2 * 3 * 18; idx += 256) {
    int c   = idx / 54;
    int rem = idx % 54;
    int dy  = rem / 18;
    int col = rem % 18;
    int hh  = h + dy - 1;
    int ww  = w0 + col - 1;
    float v = 0.f;
    if (hh >= 0 && hh < HH && ww >= 0 && ww < WW)
      v = img[(size_t)c * HWN + hh * WW + ww];
    tile[c][dy][col] = v;
  }
  __syncthreads();

  int wv = threadIdx.x >> 5;
  int ln = threadIdx.x & 31;
  int lm = ln & 15;                            // A row / B-D col
  int kh = (ln >> 4) * 2;                      // k sub-offset for this half-wave
  int hi = (ln >> 4) * 8;                      // D-matrix M offset
  int nb = wv * 16;                            // out-channel base

  v8f acc;
  float bias = bc[nb + lm];
  for (int i = 0; i < 8; ++i) acc[i] = bias;

  for (int k0 = 0; k0 < 1152; k0 += 4) {
    int tap = k0 >> 7;                         // (dy*3+dx)
    int c0  = k0 & 127;
    int dy  = tap / 3, dx = tap % 3;
    v2f a, bf;
    a.x = tile[c0 + kh][dy][lm + dx];
    a.y = tile[c0 + kh + 1][dy][lm + dx];
    const float* wp = Wc + (size_t)(k0 + kh) * C2 + nb + lm;
    bf.x = wp[0];
    bf.y = wp[C2];
    acc = WMMA_F32(a, bf, acc);
  }

  int cfull = nb + lm;                         // 0..127 (uniform half per wave)
  int cd    = cfull & 63;
  bool isIm = (cfull >= DD);
  const float* xorig = (isIm ? x_im : x_re) + ((size_t)bt * DD + cd) * HWN + h * WW + w0;
  float* outp = (isIm ? xnew_im : xnew_re) + ((size_t)bt * HWN + h * WW + w0) * DD + cd;
  for (int r = 0; r < 8; ++r) {
    int m = hi + r;
    outp[(size_t)m * DD] = acc[r] + xorig[m];
  }
}

// ---------------------------------------------------------------------------
// Kernel 3: complex GEMM  O = A * (B_re + i B_im),  A channel-last (M,64).
// 8 waves: waves 0-3 -> O_re N-tiles, waves 4-7 -> O_im N-tiles.
// O_re = Ar*Br - Ai*Bi ; O_im = Ar*Bi + Ai*Br   (32 WMMAs / wave)
// ---------------------------------------------------------------------------
__global__ void k_cgemm(const float* __restrict__ A_re, const float* __restrict__ A_im,
                        const float* __restrict__ B_re, const float* __restrict__ B_im,
                        float* __restrict__ O_re,       float* __restrict__ O_im) {
  __shared__ float As[2][16][DD];
  int row0 = blockIdx.x * 16;
  for (int idx = threadIdx.x; idx < 2048; idx += 256) {
    int pl = idx >> 10, r = (idx >> 6) & 15, k = idx & 63;
    As[pl][r][k] = (pl ? A_im : A_re)[(size_t)(row0 + r) * DD + k];
  }
  __syncthreads();

  int wv = threadIdx.x >> 5;
  int ln = threadIdx.x & 31;
  int lm = ln & 15;
  int kh = (ln >> 4) * 2;
  int hi = (ln >> 4) * 8;
  int part = wv >> 2;                          // 0 = re out, 1 = im out
  int nb   = (wv & 3) * 16;

  v8f acc;
  for (int i = 0; i < 8; ++i) acc[i] = 0.f;

  for (int k0 = 0; k0 < DD; k0 += 4) {
    v2f ar, ai, b1f, b2f;
    ar.x = As[0][lm][k0 + kh];  ar.y = As[0][lm][k0 + kh + 1];
    ai.x = As[1][lm][k0 + kh];  ai.y = As[1][lm][k0 + kh + 1];
    const float* br = B_re + (size_t)(k0 + kh) * DD + nb + lm;
    const float* bi = B_im + (size_t)(k0 + kh) * DD + nb + lm;
    if (part == 0) {                           // wave-uniform branch (EXEC stays full)
      b1f.x = br[0];   b1f.y = br[DD];
      b2f.x = -bi[0];  b2f.y = -bi[DD];
    } else {
      b1f.x = bi[0];   b1f.y = bi[DD];
      b2f.x = br[0];   b2f.y = br[DD];
    }
    acc = WMMA_F32(ar, b1f, acc);
    acc = WMMA_F32(ai, b2f, acc);
  }

  float* O = part ? O_im : O_re;
  for (int r = 0; r < 8; ++r)
    O[(size_t)(row0 + hi + r) * DD + nb + lm] = acc[r];
}

// ---------------------------------------------------------------------------
// Kernel 4a/4b: deterministic spatial mean of x_eig over H*W (as sums).
// ---------------------------------------------------------------------------
__global__ void k_reduce(const float* __restrict__ xe_re, const float* __restrict__ xe_im,
                         float* __restrict__ pr, float* __restrict__ pi) {
  int bt = blockIdx.x >> 4;                    // 16 blocks per (b,t)
  int p0 = (blockIdx.x & 15) * 256;
  int d = threadIdx.x & 63, chunk = threadIdx.x >> 6;
  float sr = 0.f, si = 0.f;
  size_t base = ((size_t)bt * HWN + p0 + chunk * 64) * DD + d;
  for (int j = 0; j < 64; ++j) {
    sr += xe_re[base + (size_t)j * DD];
    si += xe_im[base + (size_t)j * DD];
  }
  __shared__ float lr[4][64], li[4][64];
  lr[chunk][d] = sr; li[chunk][d] = si;
  __syncthreads();
  if (threadIdx.x < 64) {
    pr[(size_t)blockIdx.x * 64 + d] = lr[0][d] + lr[1][d] + lr[2][d] + lr[3][d];
    pi[(size_t)blockIdx.x * 64 + d] = li[0][d] + li[1][d] + li[2][d] + li[3][d];
  }
}

__global__ void k_reduce2(const float* __restrict__ pr, const float* __restrict__ pi,
                          float* __restrict__ sm_re, float* __restrict__ sm_im) {
  int i = blockIdx.x * 256 + threadIdx.x;      // bt*64+d, 2048 total
  if (i >= BB * TT * DD) return;
  int bt = i >> 6, d = i & 63;
  float a = 0.f, b = 0.f;
  for (int j = 0; j < 16; ++j) {
    a += pr[(size_t)(bt * 16 + j) * 64 + d];
    b += pi[(size_t)(bt * 16 + j) * 64 + d];
  }
  sm_re[i] = a; sm_im[i] = b;
}

// ---------------------------------------------------------------------------
// Kernel 5: sequence-level scan.  1 block, 128 threads = (b,d).
// flux recurrence, source matvec via LDS, gate, op_decay/op_forcing.
// ---------------------------------------------------------------------------
__global__ void k_seq(const float* __restrict__ sm_re, const float* __restrict__ sm_im,
                      const float* __restrict__ dt_seq,
                      const float* __restrict__ flux_re0, const float* __restrict__ flux_im0,
                      const float* __restrict__ lam_flux_raw,
                      const float* __restrict__ Wsrc_re, const float* __restrict__ Wsrc_im,
                      const float* __restrict__ w_gate,  const float* __restrict__ b_gate,
                      const float* __restrict__ lam_re_raw, const float* __restrict__ lam_im,
                      float* __restrict__ gate,
                      float* __restrict__ src_re, float* __restrict__ src_im,
                      float* __restrict__ od_re,  float* __restrict__ od_im,
                      float* __restrict__ of_re,  float* __restrict__ of_im,
                      float* __restrict__ flux_out) {
  __shared__ float fr[128], fi[128];
  int tid = threadIdx.x;
  int b = tid >> 6, d = tid & 63;
  float lamf = log1pf(expf(lam_flux_raw[d]));          // softplus
  float lr   = -log1pf(expf(lam_re_raw[d]));           // MAX_GROWTH - softplus
  float li   = lam_im[d];
  float den  = lr * lr + li * li;
  float flr = flux_re0[b * DD + d], fli = flux_im0[b * DD + d];
  for (int t = 0; t < TT; ++t) {
    float dt = dt_seq[b * TT + t];
    float Af = expf(-lamf * dt);
    int btd = (b * TT + t) * DD + d;
    float xmr = sm_re[btd] * (1.f / (float)HWN);
    float xmi = sm_im[btd] * (1.f / (float)HWN);
    flr = Af * flr + xmr * dt;
    fli = Af * fli + xmi * dt;
    fr[tid] = flr; fi[tid] = fli;
    __syncthreads();
    float sr = 0.f, si = 0.f;                          // source = flux @ Wsrc (complex)
    for (int k = 0; k < DD; ++k) {
      float a  = fr[b * DD + k], c = fi[b * DD + k];
      float wr = Wsrc_re[k * DD + d], wi = Wsrc_im[k * DD + d];
      sr += a * wr - c * wi;
      si += a * wi + c * wr;
    }
    float g  = 1.f / (1.f + expf(-(flr * w_gate[d] + b_gate[d])));
    float er = expf(lr * dt);
    float odr = er * cosf(li * dt), odi = er * sinf(li * dt);
    float pr = odr - 1.f;
    gate[btd]  = g;
    src_re[btd] = sr;  src_im[btd] = si;
    od_re[btd] = odr;  od_im[btd] = odi;
    of_re[btd] = (pr * lr + odi * li) / den;
    of_im[btd] = (odi * lr - pr * li) / den;
    __syncthreads();
  }
  flux_out[(b * DD + d) * 2 + 0] = flr;
  flux_out[(b * DD + d) * 2 + 1] = fli;
}

// ---------------------------------------------------------------------------
// Kernel 6: per-pixel complex recurrence over T.  u_out -> ws, h_out -> d_out.
// thread = b*2^18 + p*64 + d
// ---------------------------------------------------------------------------
__global__ void k_scan(const float* __restrict__ xe_re, const float* __restrict__ xe_im,
                       const float* __restrict__ h_re,  const float* __restrict__ h_im,
                       const float* __restrict__ gate,
                       const float* __restrict__ src_re, const float* __restrict__ src_im,
                       const float* __restrict__ od_re,  const float* __restrict__ od_im,
                       const float* __restrict__ of_re,  const float* __restrict__ of_im,
                       float* __restrict__ u_re, float* __restrict__ u_im,
                       float* __restrict__ h_out) {
  size_t idx = (size_t)blockIdx.x * 256 + threadIdx.x;   // 524288 threads
  int d = (int)(idx & 63);
  int b = (int)(idx >> 18);
  size_t pd = idx & 0x3FFFF;                             // p*64+d
  float sr = h_re[idx], si = h_im[idx];
  for (int t = 0; t < TT; ++t) {
    int btd = (b * TT + t) * DD + d;
    float g = gate[btd];
    float s_r = src_re[btd], s_i = src_im[btd];
    float odr = od_re[btd],  odi = od_im[btd];
    float ofr = of_re[btd],  ofi = of_im[btd];
    size_t xa = ((size_t)(b * TT + t) << 18) + pd;
    float xr = xe_re[xa], xi = xe_im[xa];
    float frc = xr * g + s_r * (1.f - g);
    float fic = xi * g + s_i * (1.f - g);
    float ur = frc * ofr - fic * ofi;                    // forcing * op_forcing
    float ui = frc * ofi + fic * ofr;
    float nsr = odr * sr - odi * si + ur;                // s = od*s + u
    float nsi = odr * si + odi * sr + ui;
    sr = nsr; si = nsi;
    u_re[xa] = sr; u_im[xa] = si;
  }
  h_out[idx * 2 + 0] = sr;
  h_out[idx * 2 + 1] = si;
}

// ---------------------------------------------------------------------------
// Kernel 7: fused LN + MLP(128->512 GELU ->128) + final residual sum.
//   z = x_new + decoded + delta, interleaved (.,D,2) output.
// ---------------------------------------------------------------------------
__device__ __forceinline__ float gelu_tanh(float x) {
  float x3 = x * x * x;
  float t  = tanhf(0.7978845608028654f * (x + 0.044715f * x3));
  return 0.5f * x * (1.f + t);
}

__global__ void k_mlp(const float* __restrict__ dec_re, const float* __restrict__ dec_im,
                      const float* __restrict__ xnew_re, const float* __restrict__ xnew_im,
                      const float* __restrict__ g_t, const float* __restrict__ b_t,
                      const float* __restrict__ W1,  const float* __restrict__ b1,
                      const float* __restrict__ W2,  const float* __restrict__ b2,
                      float* __restrict__ z_out) {
  __shared__ float raw[16][C2];
  __shared__ float dn[16][C2];
  __shared__ float h1s[16][512];
  __shared__ float mv[16], rv[16];
  int row0 = blockIdx.x * 16;
  int tid  = threadIdx.x;

  for (int idx = tid; idx < 16 * C2; idx += 256) {
    int r = idx >> 7, c = idx & 127;
    raw[r][c] = (c < DD) ? dec_re[(size_t)(row0 + r) * DD + c]
                         : dec_im[(size_t)(row0 + r) * DD + (c - DD)];
  }
  __syncthreads();
  if (tid < 16) {
    float s = 0.f, ss = 0.f;
    for (int c = 0; c < C2; ++c) { float v = raw[tid][c]; s += v; ss += v * v; }
    float m = s * (1.f / 128.f);
    mv[tid] = m;
    rv[tid] = rsqrtf(ss * (1.f / 128.f) - m * m + 1e-5f);
  }
  __syncthreads();
  for (int idx = tid; idx < 16 * C2; idx += 256) {
    int r = idx >> 7, c = idx & 127;
    dn[r][c] = (raw[r][c] - mv[r]) * rv[r] * g_t[c] + b_t[c];
  }
  __syncthreads();

  int wv = tid >> 5;
  int ln = tid & 31;
  int lm = ln & 15;
  int kh = (ln >> 4) * 2;
  int hi = (ln >> 4) * 8;

  // GEMM1: 16x512, K=128.  Each wave: 4 N-tiles, 32 WMMAs each.
  for (int nt = 0; nt < 4; ++nt) {
    int nb = (wv * 4 + nt) * 16;
    v8f acc;
    float bias = b1[nb + lm];
    for (int i = 0; i < 8; ++i) acc[i] = bias;
    for (int k0 = 0; k0 < C2; k0 += 4) {
      v2f a, bf;
      a.x = dn[lm][k0 + kh];  a.y = dn[lm][k0 + kh + 1];
      const float* wp = W1 + (size_t)(k0 + kh) * 512 + nb + lm;
      bf.x = wp[0];  bf.y = wp[512];
      acc = WMMA_F32(a, bf, acc);
    }
    for (int r = 0; r < 8; ++r)
      h1s[hi + r][nb + lm] = gelu_tanh(acc[r]);
  }
  __syncthreads();

  // GEMM2: 16x128, K=512.  Each wave: 1 N-tile, 128 WMMAs.
  int nb2 = wv * 16;
  v8f acc2;
  float bias2 = b2[nb2 + lm];
  for (int i = 0; i < 8; ++i) acc2[i] = bias2;
  for (int k0 = 0; k0 < 512; k0 += 4) {
    v2f a, bf;
    a.x = h1s[lm][k0 + kh];  a.y = h1s[lm][k0 + kh + 1];
    const float* wp = W2 + (size_t)(k0 + kh) * C2 + nb2 + lm;
    bf.x = wp[0];  bf.y = wp[C2];
    acc2 = WMMA_F32(a, bf, acc2);
  }

  int cfull = nb2 + lm;
  int cd    = cfull & 63;
  bool isIm = (cfull >= DD);                    // wave-uniform
  const float* xp = isIm ? xnew_im : xnew_re;
  const float* dp = isIm ? dec_im  : dec_re;
  for (int r = 0; r < 8; ++r) {
    size_t off = (size_t)(row0 + hi + r) * DD + cd;
    z_out[off * 2 + (isIm ? 1 : 0)] = acc2[r] + xp[off] + dp[off];
  }
}

// ---------------------------------------------------------------------------
// Launcher
// ---------------------------------------------------------------------------
extern "C" void kernel_launch(void* const* d_in, const int* in_sizes, int n_in,
                              void* d_out, int out_size, void* d_ws, size_t ws_size,
                              hipStream_t stream) {
  (void)in_sizes; (void)n_in; (void)out_size; (void)ws_size;
  const float* x_re   = (const float*)d_in[0];
  const float* x_im   = (const float*)d_in[1];
  const float* h_re   = (const float*)d_in[2];
  const float* h_im   = (const float*)d_in[3];
  const float* fl_re  = (const float*)d_in[4];
  const float* fl_im  = (const float*)d_in[5];
  const float* dt_seq = (const float*)d_in[6];
  const float* g_s    = (const float*)d_in[7];
  const float* b_s    = (const float*)d_in[8];
  const float* Wc     = (const float*)d_in[9];
  const float* bc     = (const float*)d_in[10];
  const float* g_t    = (const float*)d_in[11];
  const float* b_t    = (const float*)d_in[12];
  const float* E_re   = (const float*)d_in[13];
  const float* E_im   = (const float*)d_in[14];
  const float* Dec_re = (const float*)d_in[15];
  const float* Dec_im = (const float*)d_in[16];
  const float* lam_flux_raw = (const float*)d_in[17];
  const float* Wsrc_re = (const float*)d_in[18];
  const float* Wsrc_im = (const float*)d_in[19];
  const float* w_gate  = (const float*)d_in[20];
  const float* b_gate  = (const float*)d_in[21];
  const float* lam_re_raw = (const float*)d_in[22];
  const float* lam_im  = (const float*)d_in[23];
  const float* W1 = (const float*)d_in[24];
  const float* b1 = (const float*)d_in[25];
  const float* W2 = (const float*)d_in[26];
  const float* b2 = (const float*)d_in[27];

  // Workspace layout (floats).  Aliasing: u_* reuses xn; dec_* reuses x_eig.
  const size_t PLANE = (size_t)NPOS * DD;      // 8,388,608
  float* ws       = (float*)d_ws;
  float* xn       = ws;                        // (B,T,128,H,W)  [dead after conv]
  float* u_re     = ws;                        // aliases xn
  float* u_im     = ws + PLANE;
  float* xnew_re  = ws + 2 * PLANE;
  float* xnew_im  = ws + 3 * PLANE;
  float* xeig_re  = ws + 4 * PLANE;            // [dead after k_scan]
  float* xeig_im  = ws + 5 * PLANE;
  float* dec_re   = xeig_re;                   // aliases x_eig
  float* dec_im   = xeig_im;
  float* smallb   = ws + 6 * PLANE;
  float* sm_re  = smallb;          float* sm_im  = smallb + 2048;
  float* gate   = smallb + 4096;
  float* src_re = smallb + 6144;   float* src_im = smallb + 8192;
  float* od_re  = smallb + 10240;  float* od_im  = smallb + 12288;
  float* of_re  = smallb + 14336;  float* of_im  = smallb + 16384;
  float* part_re = smallb + 18432;             // 512*64 partial sums
  float* part_im = smallb + 18432 + 32768;

  float* out_z = (float*)d_out;                // (B,T,H,W,D,2) = 16,777,216
  float* out_h = out_z + (size_t)NPOS * DD * 2;// (B*H*W,1,D,2) = 1,048,576
  float* out_f = out_h + (size_t)BB * HWN * DD * 2; // (B,D,2) = 256

  // 1. LayerNorm
  k_ln<<<NPOS / 256, 256, 0, stream>>>(x_re, x_im, g_s, b_s, xn);
  // 2. 3x3 conv (implicit GEMM, WMMA) + residual
  k_conv<<<NPOS / 16, 256, 0, stream>>>(xn, Wc, bc, x_re, x_im, xnew_re, xnew_im);
  // 3. x_eig = x_new @ (E_re + i E_im)   (complex WMMA GEMM)
  k_cgemm<<<NPOS / 16, 256, 0, stream>>>(xnew_re, xnew_im, E_re, E_im, xeig_re, xeig_im);
  // 4. spatial mean (deterministic two-stage)
  k_reduce<<<BB * TT * 16, 256, 0, stream>>>(xeig_re, xeig_im, part_re, part_im);
  k_reduce2<<<8, 256, 0, stream>>>(part_re, part_im, sm_re, sm_im);
  // 5. flux scan / gate / operator coefficients
  k_seq<<<1, 128, 0, stream>>>(sm_re, sm_im, dt_seq, fl_re, fl_im, lam_flux_raw,
                               Wsrc_re, Wsrc_im, w_gate, b_gate, lam_re_raw, lam_im,
                               gate, src_re, src_im, od_re, od_im, of_re, of_im, out_f);
  // 6. complex recurrence over T (writes u_out over the dead xn region)
  k_scan<<<(BB * HWN * DD) / 256, 256, 0, stream>>>(
      xeig_re, xeig_im, h_re, h_im, gate, src_re, src_im,
      od_re, od_im, of_re, of_im, u_re, u_im, out_h);
  // 7. decoded = u_out @ (Dec_re + i Dec_im)  (writes over dead x_eig region)
  k_cgemm<<<NPOS / 16, 256, 0, stream>>>(u_re, u_im, Dec_re, Dec_im, dec_re, dec_im);
  // 8. LN + MLP + final residual -> z
  k_mlp<<<NPOS / 16, 256, 0, stream>>>(dec_re, dec_im, xnew_re, xnew_im,
                                       g_t, b_t, W1, b1, W2, b2, out_z);
}